// SparseFusionTransformer_66082366816345
// MI455X (gfx1250) — compile-verified
//
#include <hip/hip_runtime.h>

// Problem constants from the reference
#define B_DIM   8
#define D_DIM   256
#define S_DIM   4096
#define K_SEL   512
#define RCHUNKS 16
#define ROWS_PER_CHUNK (S_DIM / RCHUNKS)   // 256

typedef __attribute__((ext_vector_type(2))) float v2f;
typedef __attribute__((ext_vector_type(4))) float v4f;
typedef __attribute__((ext_vector_type(8))) float v8f;

#if defined(__gfx1250__) && __has_builtin(__builtin_amdgcn_wmma_f32_16x16x4_f32)
#define HAVE_WMMA_F32X4 1
#endif

// ---------------------------------------------------------------------------
// Phase 1: partial column sums of w.  w[b][i][c] -> partial[b][r][c]
// Streams 537 MB once; NT loads keep it out of the 192MB L2.
// grid: (4 col tiles of 1024, RCHUNKS, B), block 256, float4 per thread.
// ---------------------------------------------------------------------------
__global__ void sft_colsum_partial(const float* __restrict__ w,
                                   float* __restrict__ partial) {
    const int tile = blockIdx.x;
    const int r    = blockIdx.y;
    const int b    = blockIdx.z;
    const int c    = tile * 1024 + threadIdx.x * 4;

    const v4f* wp = reinterpret_cast<const v4f*>(
        w + (size_t)b * S_DIM * S_DIM + (size_t)r * ROWS_PER_CHUNK * S_DIM + c);

    v4f acc = {0.f, 0.f, 0.f, 0.f};
#pragma unroll 8
    for (int i = 0; i < ROWS_PER_CHUNK; ++i) {
        v4f v = __builtin_nontemporal_load(wp + (size_t)i * (S_DIM / 4));
        acc += v;
    }
    *reinterpret_cast<v4f*>(
        partial + ((size_t)(b * RCHUNKS + r)) * S_DIM + c) = acc;
}

// ---------------------------------------------------------------------------
// Phase 2: reduce 16 partials per column with V_WMMA_F32_16X16X4_F32.
// A = ones(16x4)  ->  D[m,n] = sum_k B[k,n]  (permutation-invariant in K).
// Chain 4 WMMAs to cover the 16 partial rows. One wave per 16-column group.
// ---------------------------------------------------------------------------
__global__ void sft_colsum_finalize(const float* __restrict__ partial,
                                    float* __restrict__ colsum) {
    const int wave = (blockIdx.x * blockDim.x + threadIdx.x) >> 5;
    const int lane = threadIdx.x & 31;
    const int b    = wave >> 8;          // 256 column-groups per batch
    const int cg   = wave & 255;
    const int c0   = cg * 16;
    const int n    = lane & 15;
    const int half = lane >> 4;

    const float* pb = partial + (size_t)b * RCHUNKS * S_DIM;

#if defined(HAVE_WMMA_F32X4)
    v8f acc = {};
    v2f a;  a.x = 1.0f; a.y = 1.0f;      // all-ones A: layout-invariant
#pragma unroll
    for (int kk = 0; kk < 4; ++kk) {
        const int r0 = kk * 4 + half * 2;
        v2f bf;
        bf.x = pb[(size_t)(r0 + 0) * S_DIM + c0 + n];
        bf.y = pb[(size_t)(r0 + 1) * S_DIM + c0 + n];
        acc = __builtin_amdgcn_wmma_f32_16x16x4_f32(
            /*neg_a=*/false, a, /*neg_b=*/false, bf,
            /*c_mod=*/(short)0, acc, /*reuse_a=*/false, /*reuse_b=*/false);
    }
    const float sum = acc[0];            // D VGPR0: lanes0-15 = (M=0, N=lane)
#else
    float sum = 0.f;
    for (int r = 0; r < RCHUNKS; ++r)
        sum += pb[(size_t)r * S_DIM + c0 + n];
#endif

    if (lane < 16)
        colsum[(size_t)b * S_DIM + c0 + n] = sum;   // scale by 1/S irrelevant to top-k
}

// ---------------------------------------------------------------------------
// Phase 3: per-batch deterministic top-K via full bitonic sort in LDS.
// Order: value descending, index ascending on ties (matches jax.lax.top_k).
// ---------------------------------------------------------------------------
__global__ __launch_bounds__(1024) void sft_topk_sort(
        const float* __restrict__ colsum, int* __restrict__ idx_out) {
    __shared__ float sv[S_DIM];
    __shared__ int   si[S_DIM];
    const int b = blockIdx.x;

    for (int t = threadIdx.x; t < S_DIM; t += blockDim.x) {
        sv[t] = colsum[(size_t)b * S_DIM + t];
        si[t] = t;
    }
    __syncthreads();

    for (int k = 2; k <= S_DIM; k <<= 1) {
        for (int j = k >> 1; j > 0; j >>= 1) {
            for (int t = threadIdx.x; t < S_DIM; t += blockDim.x) {
                const int ixj = t ^ j;
                if (ixj > t) {
                    const bool asc = ((t & k) == 0);
                    const float va = sv[t], vb = sv[ixj];
                    const int   ia = si[t], ib = si[ixj];
                    // "wrong" = element at ixj strictly precedes element at t
                    const bool wrong = (vb > va) || (vb == va && ib < ia);
                    if (wrong == asc) {
                        sv[t] = vb; sv[ixj] = va;
                        si[t] = ib; si[ixj] = ia;
                    }
                }
            }
            __syncthreads();
        }
    }
    for (int t = threadIdx.x; t < K_SEL; t += blockDim.x)
        idx_out[b * K_SEL + t] = si[t];
}

// ---------------------------------------------------------------------------
// Phase 4: gather  out[b][d][j] = x[b][d][idx[b][j]].  Coalesced writes;
// gathered reads hit L2 across the d-loop (idx set identical for all d).
// ---------------------------------------------------------------------------
__global__ void sft_gather(const float* __restrict__ x,
                           const int* __restrict__ idx,
                           float* __restrict__ out) {
    const int e = blockIdx.x * blockDim.x + threadIdx.x;   // < B*D*K = 2^20
    const int j = e & (K_SEL - 1);
    const int d = (e >> 9) & (D_DIM - 1);
    const int b = e >> 17;
    const int col = idx[b * K_SEL + j];
    out[e] = x[((size_t)(b * D_DIM + d)) * S_DIM + col];
}

// ---------------------------------------------------------------------------
extern "C" void kernel_launch(void* const* d_in, const int* in_sizes, int n_in,
                              void* d_out, int out_size, void* d_ws, size_t ws_size,
                              hipStream_t stream) {
    const float* x = (const float*)d_in[0];      // [B, D, S]
    const float* w = (const float*)d_in[1];      // [B, S, S]
    // d_in[2] is k (==512), baked in at compile time.
    float* out = (float*)d_out;                  // [B, D, K]

    float* partial = (float*)d_ws;                                   // B*R*S  floats (2 MB)
    float* colsum  = partial + (size_t)B_DIM * RCHUNKS * S_DIM;      // B*S    floats (128 KB)
    int*   idxb    = (int*)(colsum + (size_t)B_DIM * S_DIM);         // B*K    ints   (16 KB)

    dim3 g1(S_DIM / 1024, RCHUNKS, B_DIM);                 // 4 x 16 x 8 = 512 blocks
    sft_colsum_partial<<<g1, 256, 0, stream>>>(w, partial);

    const int waves2  = B_DIM * (S_DIM / 16);              // 2048 waves
    const int blocks2 = waves2 * 32 / 256;                 // 256 blocks
    sft_colsum_finalize<<<blocks2, 256, 0, stream>>>(partial, colsum);

    sft_topk_sort<<<B_DIM, 1024, 0, stream>>>(colsum, idxb);

    const int elems = B_DIM * D_DIM * K_SEL;               // 1M
    sft_gather<<<elems / 256, 256, 0, stream>>>(x, idxb, out);
}